// DMRInduction_12713103197293
// MI455X (gfx1250) — compile-verified
//
#include <hip/hip_runtime.h>
#include <math.h>

// Problem constants (match reference)
#define BATCH 256
#define NCAP  16
#define ICAP  128
#define DCAP  64
#define DIN   768
#define NID   (NCAP*ICAP*DCAP)    // 131072  rows of the big GEMM
#define BNI   (BATCH*NCAP*ICAP)   // 524288
#define EPSF  1e-8f
#define NCHUNK (DIN/32)           // 24 K-chunks

typedef float v2f __attribute__((ext_vector_type(2)));
typedef float v8f __attribute__((ext_vector_type(8)));
typedef int   v4i __attribute__((ext_vector_type(4)));

// CDNA5 async global->LDS path (ASYNCcnt), guarded so absence degrades to sync staging
#if __has_builtin(__builtin_amdgcn_global_load_async_to_lds_b128) && \
    __has_builtin(__builtin_amdgcn_s_wait_asynccnt)
#define HAVE_ASYNC_LDS 1
typedef __attribute__((address_space(1))) v4i* gptr_b128;   // global int4*
typedef __attribute__((address_space(3))) v4i* lptr_b128;   // LDS int4*
#else
#define HAVE_ASYNC_LDS 0
#endif

// ---------------------------------------------------------------------------
// Pair-interleaved transpose: qP[kp][b][r] = q[b][2*kp + r]
// A WMMA B-fragment (two consecutive K for this lane) becomes ONE b64 load.
// ---------------------------------------------------------------------------
__global__ void qP_kernel(const float* __restrict__ q, float* __restrict__ qP) {
  int idx = blockIdx.x * 256 + threadIdx.x;   // 0 .. 196607
  int kp = idx >> 9;                          // K pair index (0..383)
  int rest = idx & 511;
  int b = rest >> 1, r = rest & 1;
  qP[idx] = q[b * DIN + (kp << 1) + r];
}

// ---------------------------------------------------------------------------
// Fused: hat_q[b,c,i,p] = sum_e W[c,i,p,e] q[b,e]          (WMMA f32 16x16x4)
//        hat_m[c,i,p]   = sum_e W[c,i,p,e] m[i,e] + bias   (rides the same LDS tile)
// One block per (c,i). 8 waves; each wave: 4 M-strips x 2 N-tiles.
// W chunks (64x32) double-buffered in LDS via async-to-LDS b128 when available.
// ---------------------------------------------------------------------------
#define WPAD 36   // 32 K-cols + 4 pad: 16B-aligned rows for b128, conflict-free A-frags

__device__ __forceinline__ void stage_W(const float* __restrict__ Wci, int k0,
                                        float* Wbuf, int tid) {
#if HAVE_ASYNC_LDS
  #pragma unroll
  for (int j = 0; j < 2; j++) {
    int g = tid + 256 * j;                 // 0..511 b128 transfers
    int row = g >> 3, seg = (g & 7) << 2;  // seg in floats (16B granules)
    __builtin_amdgcn_global_load_async_to_lds_b128(
        (gptr_b128)(Wci + row * DIN + k0 + seg),
        (lptr_b128)(Wbuf + row * WPAD + seg), 0, 0);
  }
#else
  #pragma unroll
  for (int j = 0; j < 8; j++) {
    int l = tid + 256 * j;
    int r = l >> 5, c = l & 31;
    Wbuf[r * WPAD + c] = Wci[r * DIN + k0 + c];
  }
  if (k0 + 32 < DIN)   // software prefetch of the chunk after next
    __builtin_prefetch(&Wci[(tid >> 2) * DIN + k0 + 32 + ((tid & 3) << 3)], 0, 0);
#endif
}

__global__ void __launch_bounds__(256)
wq_wmma_kernel(const float* __restrict__ W, const float* __restrict__ qP,
               const float* __restrict__ m, const float* __restrict__ bias,
               float* __restrict__ hatq /*== tq*/, float* __restrict__ hatm) {
  __shared__ float Wl0[64 * WPAD];
  __shared__ float Wl1[64 * WPAD];
  __shared__ float ml[32];
  __shared__ float hmred[4 * 64];

  const int ci   = blockIdx.x;            // c*ICAP + i
  const int icap = ci & (ICAP - 1);
  const int tid  = threadIdx.x;
  const int w    = tid >> 5;              // wave 0..7
  const int lane = tid & 31;
  const int half = lane >> 4;             // 0: lanes 0-15, 1: lanes 16-31
  const int lr   = lane & 15;

  const float* Wci = W + (size_t)ci * DCAP * DIN;

  v8f acc[4][2] = {};                     // 16x16 f32 accum tiles
  float hm = 0.f;                         // hat_m partial (p = tid&63, seg = tid>>6)

  stage_W(Wci, 0, Wl0, tid);              // prologue: chunk 0 in flight

  for (int chunk = 0; chunk < NCHUNK; ++chunk) {
    const int k0 = chunk * 32;
    float* Wb = (chunk & 1) ? Wl1 : Wl0;

    if (tid < 32) ml[tid] = m[icap * DIN + k0 + tid];

    if (chunk + 1 < NCHUNK) {
      stage_W(Wci, k0 + 32, (chunk & 1) ? Wl0 : Wl1, tid);
#if HAVE_ASYNC_LDS
      // 2 newer (next-chunk) asyncs may remain; in-order completion =>
      // count<=2 guarantees current chunk has landed.
      __builtin_amdgcn_s_wait_asynccnt(2);
#endif
    } else {
#if HAVE_ASYNC_LDS
      __builtin_amdgcn_s_wait_asynccnt(0);
#endif
    }
    __syncthreads();   // all waves' waits done -> current chunk + ml visible

    // hat_m partials off the LDS-resident tile (8 FMAs/thread: negligible)
    {
      int pp = tid & 63, seg = tid >> 6;
      #pragma unroll
      for (int u = 0; u < 8; u++)
        hm += Wb[pp * WPAD + seg * 8 + u] * ml[seg * 8 + u];
    }

#if __has_builtin(__builtin_amdgcn_wmma_f32_16x16x4_f32)
    #pragma unroll
    for (int kk = 0; kk < 32; kk += 4) {
      const int k = k0 + kk;
      // A frag (16x4 f32): lanes0-15 K={0,1}, lanes16-31 K={2,3}   (ISA 7.12.2)
      v2f af[4];
      #pragma unroll
      for (int s = 0; s < 4; s++) {
        af[s][0] = Wb[(16 * s + lr) * WPAD + kk + 2 * half];
        af[s][1] = Wb[(16 * s + lr) * WPAD + kk + 2 * half + 1];
      }
      // B frag: one contiguous b64 per tile from pair-interleaved qP
      v2f bf[2];
      #pragma unroll
      for (int t = 0; t < 2; t++) {
        int n = w * 32 + 16 * t + lr;
        int kp = (k >> 1) + half;
        bf[t] = *(const v2f*)(qP + ((size_t)kp << 9) + (n << 1));
      }
      #pragma unroll
      for (int s = 0; s < 4; s++)
        #pragma unroll
        for (int t = 0; t < 2; t++)
          acc[s][t] = __builtin_amdgcn_wmma_f32_16x16x4_f32(
              false, af[s], false, bf[t], (short)0, acc[s][t], false, false);
    }
#else
    // Scalar fallback with identical lane->C mapping (keeps file compiling)
    for (int kk = 0; kk < 32; kk++) {
      const int k = k0 + kk;
      #pragma unroll
      for (int t = 0; t < 2; t++) {
        int n = w * 32 + 16 * t + lr;
        float bq = qP[((size_t)(k >> 1) << 9) + (n << 1) + (k & 1)];
        #pragma unroll
        for (int s = 0; s < 4; s++) {
          float* accf = (float*)&acc[s][t];
          #pragma unroll
          for (int j = 0; j < 8; j++)
            accf[j] += Wb[(16 * s + 8 * half + j) * WPAD + kk] * bq;
        }
      }
    }
#endif
    __syncthreads();   // everyone done reading Wb before it is re-staged
  }

  // Epilogue: lane owns C rows M = j + 8*half -> 8 CONTIGUOUS p values
  #pragma unroll
  for (int s = 0; s < 4; s++)
    #pragma unroll
    for (int t = 0; t < 2; t++) {
      float* accf = (float*)&acc[s][t];
      size_t col = (size_t)(w * 32 + 16 * t + lr);                 // batch b
      float* dst = hatq + col * NID + (size_t)ci * DCAP + 16 * s + 8 * half;
      ((float4*)dst)[0] = make_float4(accf[0], accf[1], accf[2], accf[3]);
      ((float4*)dst)[1] = make_float4(accf[4], accf[5], accf[6], accf[7]);
    }

  hmred[(tid >> 6) * 64 + (tid & 63)] = hm;
  __syncthreads();
  if (tid < 64) {
    float s = hmred[tid] + hmred[64 + tid] + hmred[128 + tid] + hmred[192 + tid];
    hatm[ci * DCAP + tid] = s + bias[ci * DCAP + tid];
  }
}

// ---------------------------------------------------------------------------
// p = tanh(-pearsonr(hat_m, tq)) over DC; also a = 0  (one thread per (b,c,i))
// ---------------------------------------------------------------------------
__global__ void pearson_init_kernel(const float* __restrict__ hatm,
                                    const float* __restrict__ tq,
                                    float* __restrict__ p, float* __restrict__ a) {
  int idx = blockIdx.x * 256 + threadIdx.x;   // BNI
  int b = idx >> 11;                          // / (NC*IC)
  int r = idx & 2047;                         // c*IC + i
  const float4* x4 = (const float4*)(hatm + r * DCAP);
  const float4* y4 = (const float4*)(tq + (size_t)b * NID + r * DCAP);
  float sx = 0, sy = 0, sxy = 0, sxx = 0, syy = 0;
  #pragma unroll
  for (int j = 0; j < 16; j++) {
    float4 x = x4[j], y = y4[j];
    sx  += x.x + x.y + x.z + x.w;
    sy  += y.x + y.y + y.z + y.w;
    sxy += x.x * y.x + x.y * y.y + x.z * y.z + x.w * y.w;
    sxx += x.x * x.x + x.y * x.y + x.z * x.z + x.w * x.w;
    syy += y.x * y.x + y.y * y.y + y.z * y.z + y.w * y.w;
  }
  const float invn = 1.f / 64.f;
  float num = sxy - sx * sy * invn;
  float vx  = fmaxf(sxx - sx * sx * invn, 0.f);
  float vy  = fmaxf(syy - sy * sy * invn, 0.f);
  float corr = num / (sqrtf(vx) * sqrtf(vy) + EPSF);
  p[idx] = tanhf(-corr);
  a[idx] = 0.f;
}

// ---------------------------------------------------------------------------
// d = softmax(a, axis=NC)   (one thread per (b,i))
// ---------------------------------------------------------------------------
__global__ void softmax_kernel(const float* __restrict__ a, float* __restrict__ d) {
  int idx = blockIdx.x * 256 + threadIdx.x;   // B*IC
  int b = idx >> 7, i = idx & 127;
  int base = b * NCAP * ICAP + i;
  float av[16], mx = -1e30f;
  #pragma unroll
  for (int c = 0; c < 16; c++) { av[c] = a[base + c * ICAP]; mx = fmaxf(mx, av[c]); }
  float s = 0.f;
  #pragma unroll
  for (int c = 0; c < 16; c++) { av[c] = expf(av[c] - mx); s += av[c]; }
  float inv = 1.f / s;
  #pragma unroll
  for (int c = 0; c < 16; c++) d[base + c * ICAP] = av[c] * inv;
}

// ---------------------------------------------------------------------------
// hv = sum_i (d+p)*hat_m ; vout = squash(hv)   (one 64-thread block per (b,c))
// Also serves as the final output kernel (vout layout == [B, NC*DC] flat).
// ---------------------------------------------------------------------------
__global__ void hv_squash_kernel(const float* __restrict__ d, const float* __restrict__ p,
                                 const float* __restrict__ hatm, float* __restrict__ vout) {
  __shared__ float red[2];
  int bc = blockIdx.x;            // b*16 + c
  int c  = bc & 15;
  int pp = threadIdx.x;           // 0..63
  const float* drow = d + bc * ICAP;
  const float* prow = p + bc * ICAP;
  float hv = 0.f;
  for (int i = 0; i < ICAP; i++) {
    float wgt = drow[i] + prow[i];
    hv += wgt * hatm[(c * ICAP + i) * DCAP + pp];
  }
  float t = hv * hv;                                   // wave32 reduction
  for (int off = 16; off; off >>= 1) t += __shfl_xor(t, off, 32);
  if ((threadIdx.x & 31) == 0) red[threadIdx.x >> 5] = t;
  __syncthreads();
  float n2 = red[0] + red[1];
  vout[bc * DCAP + pp] = hv * n2 / ((1.f + n2) * sqrtf(n2 + EPSF));
}

// ---------------------------------------------------------------------------
// Fused routing update (one thread per (b,c,i)):
//   mcv = <hat_m, v>; a += p*mcv; tq = (tq+v)*0.5 in place; p = tanh(-pearson)
// ---------------------------------------------------------------------------
__global__ void update_kernel(const float* __restrict__ hatm, float* __restrict__ tq,
                              const float* __restrict__ v, float* __restrict__ a,
                              float* __restrict__ p) {
  int idx = blockIdx.x * 256 + threadIdx.x;   // BNI
  int b = idx >> 11;
  int r = idx & 2047;                         // c*IC + i
  int bc = (b << 4) | (r >> 7);               // b*16 + c
  const float4* x4 = (const float4*)(hatm + r * DCAP);
  float4*       y4 = (float4*)(tq + (size_t)b * NID + r * DCAP);
  const float4* v4 = (const float4*)(v + bc * DCAP);
  float mcv = 0, sx = 0, sy = 0, sxy = 0, sxx = 0, syy = 0;
  #pragma unroll
  for (int j = 0; j < 16; j++) {
    float4 x = x4[j], y = y4[j], vv = v4[j];
    mcv += x.x * vv.x + x.y * vv.y + x.z * vv.z + x.w * vv.w;
    float4 yn = make_float4((y.x + vv.x) * 0.5f, (y.y + vv.y) * 0.5f,
                            (y.z + vv.z) * 0.5f, (y.w + vv.w) * 0.5f);
    y4[j] = yn;
    sx  += x.x + x.y + x.z + x.w;
    sy  += yn.x + yn.y + yn.z + yn.w;
    sxy += x.x * yn.x + x.y * yn.y + x.z * yn.z + x.w * yn.w;
    sxx += x.x * x.x + x.y * x.y + x.z * x.z + x.w * x.w;
    syy += yn.x * yn.x + yn.y * yn.y + yn.z * yn.z + yn.w * yn.w;
  }
  a[idx] += p[idx] * mcv;
  const float invn = 1.f / 64.f;
  float num = sxy - sx * sy * invn;
  float vx  = fmaxf(sxx - sx * sx * invn, 0.f);
  float vy  = fmaxf(syy - sy * sy * invn, 0.f);
  float corr = num / (sqrtf(vx) * sqrtf(vy) + EPSF);
  p[idx] = tanhf(-corr);
}

// ---------------------------------------------------------------------------
extern "C" void kernel_launch(void* const* d_in, const int* in_sizes, int n_in,
                              void* d_out, int out_size, void* d_ws, size_t ws_size,
                              hipStream_t stream) {
  (void)in_sizes; (void)n_in; (void)out_size; (void)ws_size;
  const float* m    = (const float*)d_in[0];   // [128,768]
  const float* q    = (const float*)d_in[1];   // [256,768]
  const float* W    = (const float*)d_in[2];   // [16,128,64,768]
  const float* bias = (const float*)d_in[3];   // [16,128,64]
  float* out = (float*)d_out;                  // [256, 1024]

  float* qP   = (float*)d_ws;                  // 196608
  float* hatm = qP   + 196608;                 // 131072
  float* tq   = hatm + 131072;                 // 33554432 (hat_q, updated in place)
  float* a    = tq   + 33554432;               // 524288
  float* p    = a    + 524288;                 // 524288
  float* dd   = p    + 524288;                 // 524288
  float* v    = dd   + 524288;                 // 262144

  qP_kernel<<<768, 256, 0, stream>>>(q, qP);
  wq_wmma_kernel<<<NCAP * ICAP, 256, 0, stream>>>(W, qP, m, bias, tq, hatm);
  pearson_init_kernel<<<BNI / 256, 256, 0, stream>>>(hatm, tq, p, a);

  for (int it = 0; it < 2; ++it) {             // NUM_ROUTING - 1
    softmax_kernel<<<(BATCH * ICAP) / 256, 256, 0, stream>>>(a, dd);
    hv_squash_kernel<<<BATCH * NCAP, DCAP, 0, stream>>>(dd, p, hatm, v);
    update_kernel<<<BNI / 256, 256, 0, stream>>>(hatm, tq, v, a, p);
  }
  softmax_kernel<<<(BATCH * ICAP) / 256, 256, 0, stream>>>(a, dd);
  hv_squash_kernel<<<BATCH * NCAP, DCAP, 0, stream>>>(dd, p, hatm, out);
}